// DropEdgeModel_17222818857596
// MI455X (gfx1250) — compile-verified
//
#include <hip/hip_runtime.h>

typedef __attribute__((ext_vector_type(2))) float v2f;
typedef __attribute__((ext_vector_type(8))) float v8f;

#define FIN 512
#define HID 16
#define FOUT 40

// ---------------------------------------------------------------------------
// Degree / normalization
// ---------------------------------------------------------------------------
__global__ void deg_init_kernel(float* __restrict__ deg, int N) {
  int i = blockIdx.x * blockDim.x + threadIdx.x;
  if (i < N) deg[i] = 1.0f;  // self loop contributes 1 to every node's degree
}

__global__ void deg_accum_kernel(const long long* __restrict__ ei, float* __restrict__ deg,
                                 int E) {
  int e = blockIdx.x * blockDim.x + threadIdx.x;
  if (e >= E) return;
  int c = (int)ei[(size_t)E + e];  // col = target
  unsafeAtomicAdd(&deg[c], 1.0f);
}

__global__ void dis_kernel(float* __restrict__ deg, int N) {
  int i = blockIdx.x * blockDim.x + threadIdx.x;
  if (i < N) deg[i] = __frsqrt_rn(deg[i]);  // deg >= 1 always (self loops)
}

// ---------------------------------------------------------------------------
// GEMM1: xw1[N,16] = x[N,512] @ W1[512,16] with V_WMMA_F32_16X16X4_F32
// One wave per 16-row tile; 8 waves / block; W1 staged in LDS (32 KB).
// ---------------------------------------------------------------------------
__global__ __launch_bounds__(256) void gemm1_wmma_kernel(const float* __restrict__ x,
                                                         const float* __restrict__ W1,
                                                         float* __restrict__ xw1,
                                                         int nTiles) {
  __shared__ float ldsW[FIN * HID];  // 8192 floats = 32 KB
  {
    const float4* src = (const float4*)W1;
    float4* dst = (float4*)ldsW;
    for (int i = threadIdx.x; i < (FIN * HID) / 4; i += 256) dst[i] = src[i];
  }
  __syncthreads();

  const int lane = threadIdx.x & 31;
  const int wave = threadIdx.x >> 5;
  const int tile = blockIdx.x * 8 + wave;
  if (tile >= nTiles) return;

  const int m = lane & 15;            // row within tile (A: M index, also B/C: N index)
  const int khalf = (lane >> 4) << 1; // lanes 16-31 hold K = +2,+3
  const float* xrow = x + (size_t)(tile * 16 + m) * FIN + khalf;

  v8f acc = {};
#pragma unroll 4
  for (int k = 0; k < FIN; k += 4) {
    float2 a2 = *(const float2*)(xrow + k);      // A VGPR0 = K=k+khalf, VGPR1 = K=k+khalf+1
    v2f A;
    A[0] = a2.x;
    A[1] = a2.y;
    v2f B;                                        // B VGPR0: K=k+khalf row; VGPR1: K=k+khalf+1
    B[0] = ldsW[(k + khalf) * HID + m];
    B[1] = ldsW[(k + khalf + 1) * HID + m];
    acc = __builtin_amdgcn_wmma_f32_16x16x4_f32(false, A, false, B, (short)0, acc, false, false);
  }

  // C/D layout: VGPR r -> (M = r + 8*(lane>>4), N = lane&15)
  const int mbase = (lane >> 4) << 3;
  float* outp = xw1 + (size_t)(tile * 16 + mbase) * HID + m;
#pragma unroll
  for (int r = 0; r < 8; r++) outp[(size_t)r * HID] = acc[r];
}

// ---------------------------------------------------------------------------
// Layer-1 aggregation
// ---------------------------------------------------------------------------
__global__ void agg1_init_kernel(const float* __restrict__ dis, const float* __restrict__ xw1,
                                 float* __restrict__ out1, int N) {
  int t = blockIdx.x * blockDim.x + threadIdx.x;
  if (t >= N * HID) return;
  int i = t >> 4;  // /16
  float d = dis[i];
  out1[t] = d * d * xw1[t];  // self-loop term, norm = dis[i]^2
}

__global__ void scatter1_kernel(const long long* __restrict__ ei, const float* __restrict__ dis,
                                const float* __restrict__ xw1, float* __restrict__ out1, int E) {
  int e = blockIdx.x * blockDim.x + threadIdx.x;
  if (e >= E) return;
  int r = (int)ei[e];
  int c = (int)ei[(size_t)E + e];
  float norm = dis[r] * dis[c];
  const float* src = xw1 + (size_t)r * HID;
  float* dst = out1 + (size_t)c * HID;
#pragma unroll
  for (int f = 0; f < HID; f++) unsafeAtomicAdd(&dst[f], norm * src[f]);
}

__global__ void relu_bias_kernel(float* __restrict__ h, const float* __restrict__ b1, int N) {
  int t = blockIdx.x * blockDim.x + threadIdx.x;
  if (t >= N * HID) return;
  float v = h[t] + b1[t & (HID - 1)];
  h[t] = v > 0.0f ? v : 0.0f;
}

// ---------------------------------------------------------------------------
// GEMM2: hw2[N,40] = h[N,16] @ W2[16,40]  (64 MFLOP -> scalar is fine)
// ---------------------------------------------------------------------------
__global__ void gemm2_kernel(const float* __restrict__ h, const float* __restrict__ W2,
                             float* __restrict__ hw2, int N) {
  int t = blockIdx.x * blockDim.x + threadIdx.x;
  if (t >= N * FOUT) return;
  int i = t / FOUT;
  int j = t - i * FOUT;
  const float* hr = h + (size_t)i * HID;
  float s = 0.0f;
#pragma unroll
  for (int k = 0; k < HID; k++) s = fmaf(hr[k], W2[k * FOUT + j], s);
  hw2[t] = s;
}

// ---------------------------------------------------------------------------
// Layer-2 aggregation (into d_out), bias folded into self-loop init
// ---------------------------------------------------------------------------
__global__ void agg2_init_kernel(const float* __restrict__ dis, const float* __restrict__ hw2,
                                 const float* __restrict__ b2, float* __restrict__ out, int N) {
  int t = blockIdx.x * blockDim.x + threadIdx.x;
  if (t >= N * FOUT) return;
  int i = t / FOUT;
  int j = t - i * FOUT;
  float d = dis[i];
  out[t] = d * d * hw2[t] + b2[j];
}

__global__ void scatter2_kernel(const long long* __restrict__ ei, const float* __restrict__ dis,
                                const float* __restrict__ hw2, float* __restrict__ out, int E) {
  int e = blockIdx.x * blockDim.x + threadIdx.x;
  if (e >= E) return;
  int r = (int)ei[e];
  int c = (int)ei[(size_t)E + e];
  float norm = dis[r] * dis[c];
  const float* src = hw2 + (size_t)r * FOUT;
  float* dst = out + (size_t)c * FOUT;
#pragma unroll
  for (int f = 0; f < FOUT; f++) unsafeAtomicAdd(&dst[f], norm * src[f]);
}

// ---------------------------------------------------------------------------
// Launch
// ---------------------------------------------------------------------------
extern "C" void kernel_launch(void* const* d_in, const int* in_sizes, int n_in,
                              void* d_out, int out_size, void* d_ws, size_t ws_size,
                              hipStream_t stream) {
  const float* x = (const float*)d_in[0];
  const long long* ei = (const long long*)d_in[1];
  const float* W1 = (const float*)d_in[2];
  const float* b1 = (const float*)d_in[3];
  const float* W2 = (const float*)d_in[4];
  const float* b2 = (const float*)d_in[5];
  float* out = (float*)d_out;

  const int N = in_sizes[0] / FIN;       // 100000
  const int E = in_sizes[1] / 2;         // 3200000

  // Workspace layout (floats): dis[N] | xw1[N*16] | out1/h[N*16] | hw2[N*40]  (~29.2 MB)
  float* dis = (float*)d_ws;
  float* xw1 = dis + N;
  float* h = xw1 + (size_t)N * HID;
  float* hw2 = h + (size_t)N * HID;

  const int B = 256;
  auto cdiv = [](int a, int b) { return (a + b - 1) / b; };

  // Degree / normalization (also needed before agg inits)
  deg_init_kernel<<<cdiv(N, B), B, 0, stream>>>(dis, N);
  deg_accum_kernel<<<cdiv(E, B), B, 0, stream>>>(ei, dis, E);
  dis_kernel<<<cdiv(N, B), B, 0, stream>>>(dis, N);

  // Layer 1
  const int nTiles = N / 16;  // N = 100000 is a multiple of 16
  gemm1_wmma_kernel<<<cdiv(nTiles, 8), 256, 0, stream>>>(x, W1, xw1, nTiles);
  agg1_init_kernel<<<cdiv(N * HID, B), B, 0, stream>>>(dis, xw1, h, N);
  scatter1_kernel<<<cdiv(E, B), B, 0, stream>>>(ei, dis, xw1, h, E);
  relu_bias_kernel<<<cdiv(N * HID, B), B, 0, stream>>>(h, b1, N);

  // Layer 2
  gemm2_kernel<<<cdiv(N * FOUT, B), B, 0, stream>>>(h, W2, hw2, N);
  agg2_init_kernel<<<cdiv(N * FOUT, B), B, 0, stream>>>(dis, hw2, b2, out, N);
  scatter2_kernel<<<cdiv(E, B), B, 0, stream>>>(ei, dis, hw2, out, E);
}